// IOC_11811160064289
// MI455X (gfx1250) — compile-verified
//
#include <hip/hip_runtime.h>
#include <float.h>

// ---------------------------------------------------------------------------
// Problem constants (from reference)
// ---------------------------------------------------------------------------
#define NQ     12
#define N_COL  100
#define NV     (3 * NQ * N_COL + 2 * NQ)   // 3624
#define NEQ    (2 * NQ * N_COL)            // 2400
#define NSYS   (NV + NEQ)                  // 6024
#define EPSV   0.1f

// LU blocking: pad system to a multiple of NB (and 32 for WMMA macro-tiles)
#define NB     32
#define NPAD   6048                         // 189 * 32
#define NBLK   (NPAD / NB)                  // 189

typedef float v2f __attribute__((ext_vector_type(2)));
typedef float v8f __attribute__((ext_vector_type(8)));

// ---------------------------------------------------------------------------
// CDNA5 FP32 WMMA: D(16x16,f32) = A(16x4,f32) * B(4x16,f32) + C
// ---------------------------------------------------------------------------
__device__ __forceinline__ v8f wmma_f32x4(v2f a, v2f b, v8f c) {
  return __builtin_amdgcn_wmma_f32_16x16x4_f32(
      /*neg_a=*/false, a, /*neg_b=*/false, b,
      /*c_mod=*/(short)0, c, /*reuse_a=*/false, /*reuse_b=*/false);
}

// Single-instruction relu: v_med3_f32(x, 0, FLT_MAX)
__device__ __forceinline__ float relu1(float x) {
  return __builtin_amdgcn_fmed3f(x, 0.0f, FLT_MAX);
}

// ---------------------------------------------------------------------------
// Workspace zero (matrix + rhs)
// ---------------------------------------------------------------------------
__global__ void k_zero(float* __restrict__ p, size_t n) {
  size_t i = (size_t)blockIdx.x * blockDim.x + threadIdx.x;
  if (i < n) p[i] = 0.0f;
}

// ---------------------------------------------------------------------------
// Q = 0.5 * relu(W) relu(W)^T + EPS*I  into M[0:NV,0:NV].
// One wave computes a 32x32 output region as 2x2 WMMA tiles (operand reuse 2x).
// Out-of-range lanes read a CLAMPED valid row of W: a lane's A-row only feeds
// C rows of the same index (and B-row only C cols), which the guarded store
// discards -> loads are unconditional, no EXEC-mask branches, WMMA sees
// EXEC=all-ones throughout.
//   A lane map: a[v] = A[lane%16][2*(lane/16)+v]
//   B lane map: b[v] = B[2*(lane/16)+v][lane%16]   (B = relu(W)^T)
//   C lane map: c[r] = C[r + 8*(lane/16)][lane%16]
// ---------------------------------------------------------------------------
__global__ void k_build_q(const float* __restrict__ W, float* __restrict__ M) {
  const int lane = threadIdx.x;          // 0..31
  const int half = lane >> 4;            // 0 or 1
  const int l16  = lane & 15;
  const int row0 = blockIdx.x * 32;
  const int col0 = blockIdx.y * 32;

  const int rA0 = min(row0 + l16,      NV - 1);
  const int rA1 = min(row0 + 16 + l16, NV - 1);
  const int rB0 = min(col0 + l16,      NV - 1);
  const int rB1 = min(col0 + 16 + l16, NV - 1);
  const float* wa0 = W + (size_t)rA0 * NV;
  const float* wa1 = W + (size_t)rA1 * NV;
  const float* wb0 = W + (size_t)rB0 * NV;
  const float* wb1 = W + (size_t)rB1 * NV;

  v8f acc00 = {}, acc01 = {}, acc10 = {}, acc11 = {};
  // NV = 3624 is a multiple of 4 -> exact K loop, no tail
  for (int k = 0; k < NV; k += 4) {
    const int kk = k + half * 2;
    v2f a0, a1, b0, b1;
    a0.x = relu1(wa0[kk]);  a0.y = relu1(wa0[kk + 1]);
    a1.x = relu1(wa1[kk]);  a1.y = relu1(wa1[kk + 1]);
    b0.x = relu1(wb0[kk]);  b0.y = relu1(wb0[kk + 1]);
    b1.x = relu1(wb1[kk]);  b1.y = relu1(wb1[kk + 1]);
    acc00 = wmma_f32x4(a0, b0, acc00);
    acc01 = wmma_f32x4(a0, b1, acc01);
    acc10 = wmma_f32x4(a1, b0, acc10);
    acc11 = wmma_f32x4(a1, b1, acc11);
  }

#pragma unroll
  for (int r = 0; r < 8; ++r) {
    const int ri0 = row0 + r + 8 * half;
    const int ri1 = ri0 + 16;
    const int ci0 = col0 + l16;
    const int ci1 = ci0 + 16;
    if (ri0 < NV && ci0 < NV)
      M[(size_t)ri0 * NPAD + ci0] = 0.5f * acc00[r] + (ri0 == ci0 ? EPSV : 0.0f);
    if (ri0 < NV && ci1 < NV)
      M[(size_t)ri0 * NPAD + ci1] = 0.5f * acc01[r] + (ri0 == ci1 ? EPSV : 0.0f);
    if (ri1 < NV && ci0 < NV)
      M[(size_t)ri1 * NPAD + ci0] = 0.5f * acc10[r] + (ri1 == ci0 ? EPSV : 0.0f);
    if (ri1 < NV && ci1 < NV)
      M[(size_t)ri1 * NPAD + ci1] = 0.5f * acc11[r] + (ri1 == ci1 ? EPSV : 0.0f);
  }
}

// ---------------------------------------------------------------------------
// KKT off-diagonal blocks: M[NV+r][j] = A[r][j], M[j][NV+r] = A[r][j];
// unit diagonal in the padding rows so the padded LU is nonsingular.
// ---------------------------------------------------------------------------
__global__ void k_fill_kkt(const float* __restrict__ A, float* __restrict__ M) {
  size_t t = (size_t)blockIdx.x * blockDim.x + threadIdx.x;
  const size_t nA = (size_t)NEQ * NV;
  if (t < nA) {
    const int r = (int)(t / NV);
    const int j = (int)(t % NV);
    const float v = A[t];
    M[(size_t)(NV + r) * NPAD + j] = v;        // A block
    M[(size_t)j * NPAD + (NV + r)] = v;        // A^T block
  } else if (t < nA + (size_t)(NPAD - NSYS)) {
    const int d = NSYS + (int)(t - nA);
    M[(size_t)d * NPAD + d] = 1.0f;
  }
}

// ---------------------------------------------------------------------------
// rhs = [ Q @ x_nom ; b ; 0 ]
// ---------------------------------------------------------------------------
__global__ void k_rhs(const float* __restrict__ M, const float* __restrict__ x_nom,
                      const float* __restrict__ b0, const float* __restrict__ x_init,
                      float* __restrict__ rhs) {
  const int i = blockIdx.x * blockDim.x + threadIdx.x;
  if (i < NV) {
    const float* row = M + (size_t)i * NPAD;
    float s = 0.0f;
    for (int j = 0; j < NV; ++j) s += row[j] * x_nom[j];
    rhs[i] = s;                                   // rhs_top = -q = Q x_nom
  } else if (i < NSYS) {
    const int r = i - NV;
    rhs[i] = (r >= NEQ - 2 * NQ) ? x_init[r - (NEQ - 2 * NQ)] : b0[r];
  } else if (i < NPAD) {
    rhs[i] = 0.0f;
  }
}

// ---------------------------------------------------------------------------
// Blocked right-looking LU (no pivoting; KKT ordering guarantees pivots)
// ---------------------------------------------------------------------------
__global__ void k_lu_diag(float* __restrict__ M, int k0) {
  __shared__ float T[NB][NB + 1];
  const int r = threadIdx.x;
  for (int j = 0; j < NB; ++j) T[r][j] = M[(size_t)(k0 + r) * NPAD + k0 + j];
  __syncthreads();
  for (int p = 0; p < NB; ++p) {
    if (r > p) {
      const float l = T[r][p] / T[p][p];
      T[r][p] = l;
      for (int j = p + 1; j < NB; ++j) T[r][j] -= l * T[p][j];
    }
    __syncthreads();
  }
  for (int j = 0; j < NB; ++j) M[(size_t)(k0 + r) * NPAD + k0 + j] = T[r][j];
}

// U12 = L11^-1 * A12  (unit lower L11; one thread per trailing column)
__global__ void k_trsm_left(float* __restrict__ M, int k0) {
  __shared__ float L[NB][NB];
  const int t = threadIdx.x;
  for (int idx = t; idx < NB * NB; idx += blockDim.x)
    L[idx / NB][idx % NB] = M[(size_t)(k0 + idx / NB) * NPAD + k0 + idx % NB];
  __syncthreads();
  const int col = k0 + NB + blockIdx.x * blockDim.x + t;
  if (col >= NPAD) return;
  float x[NB];
#pragma unroll
  for (int i = 0; i < NB; ++i) x[i] = M[(size_t)(k0 + i) * NPAD + col];
#pragma unroll
  for (int i = 1; i < NB; ++i) {
    float s = x[i];
    for (int p = 0; p < i; ++p) s -= L[i][p] * x[p];
    x[i] = s;
  }
#pragma unroll
  for (int i = 0; i < NB; ++i) M[(size_t)(k0 + i) * NPAD + col] = x[i];
}

// L21 = A21 * U11^-1  (one thread per trailing row)
__global__ void k_trsm_right(float* __restrict__ M, int k0) {
  __shared__ float U[NB][NB];
  const int t = threadIdx.x;
  for (int idx = t; idx < NB * NB; idx += blockDim.x)
    U[idx / NB][idx % NB] = M[(size_t)(k0 + idx / NB) * NPAD + k0 + idx % NB];
  __syncthreads();
  const int row = k0 + NB + blockIdx.x * blockDim.x + t;
  if (row >= NPAD) return;
  float* Mr = M + (size_t)row * NPAD + k0;
  float x[NB];
#pragma unroll
  for (int j = 0; j < NB; ++j) x[j] = Mr[j];
#pragma unroll
  for (int j = 0; j < NB; ++j) {
    float s = x[j];
    for (int p = 0; p < j; ++p) s -= x[p] * U[p][j];
    x[j] = s / U[j][j];
  }
#pragma unroll
  for (int j = 0; j < NB; ++j) Mr[j] = x[j];
}

// Trailing update A22 -= L21 * U12 via f32 WMMA.
// One wave computes a 32x32 region (2x2 WMMA tiles, operand reuse 2x).
// Trailing dims are exact multiples of 32 -> no guards, EXEC all ones.
__global__ void k_lu_gemm(float* __restrict__ M, int k0) {
  const int lane = threadIdx.x;
  const int half = lane >> 4;
  const int l16  = lane & 15;
  const int base = k0 + NB;
  const int row0 = base + blockIdx.x * 32;
  const int col0 = base + blockIdx.y * 32;

  const float* A0 = M + (size_t)(row0 + l16) * NPAD + k0;            // L21 rows 0..15
  const float* A1 = A0 + (size_t)16 * NPAD;                          // rows 16..31
  const float* B0 = M + (size_t)(k0 + 2 * half) * NPAD + col0 + l16; // U12 cols 0..15
  const float* B1 = B0 + 16;                                         // cols 16..31

  v8f acc00 = {}, acc01 = {}, acc10 = {}, acc11 = {};
#pragma unroll
  for (int k = 0; k < NB; k += 4) {
    const int ka = k + 2 * half;
    v2f a0, a1, b0, b1;
    a0.x = A0[ka]; a0.y = A0[ka + 1];
    a1.x = A1[ka]; a1.y = A1[ka + 1];
    b0.x = B0[(size_t)k * NPAD]; b0.y = B0[(size_t)(k + 1) * NPAD];
    b1.x = B1[(size_t)k * NPAD]; b1.y = B1[(size_t)(k + 1) * NPAD];
    acc00 = wmma_f32x4(a0, b0, acc00);
    acc01 = wmma_f32x4(a0, b1, acc01);
    acc10 = wmma_f32x4(a1, b0, acc10);
    acc11 = wmma_f32x4(a1, b1, acc11);
  }
  // f32 WMMA has no A/B negate (NEG = CNeg only) -> subtract at the RMW store
  float* C0 = M + (size_t)(row0 + 8 * half) * NPAD + col0 + l16;
  float* C1 = M + (size_t)(row0 + 16 + 8 * half) * NPAD + col0 + l16;
#pragma unroll
  for (int r = 0; r < 8; ++r) {
    C0[(size_t)r * NPAD]      -= acc00[r];
    C0[(size_t)r * NPAD + 16] -= acc01[r];
    C1[(size_t)r * NPAD]      -= acc10[r];
    C1[(size_t)r * NPAD + 16] -= acc11[r];
  }
}

// ---------------------------------------------------------------------------
// Blocked triangular solves with the factored M (L unit-lower, U upper)
// ---------------------------------------------------------------------------
__global__ void k_fwd_diag(const float* __restrict__ M, float* __restrict__ rhs, int k0) {
  __shared__ float y[NB];
  const int r = threadIdx.x;
  y[r] = rhs[k0 + r];
  __syncthreads();
  for (int p = 0; p < NB - 1; ++p) {
    if (r > p) y[r] -= M[(size_t)(k0 + r) * NPAD + k0 + p] * y[p];
    __syncthreads();
  }
  rhs[k0 + r] = y[r];
}

__global__ void k_fwd_update(const float* __restrict__ M, float* __restrict__ rhs, int k0) {
  __shared__ float y[NB];
  const int t = threadIdx.x;
  if (t < NB) y[t] = rhs[k0 + t];
  __syncthreads();
  const int i = k0 + NB + blockIdx.x * blockDim.x + t;
  if (i >= NPAD) return;
  const float* Mi = M + (size_t)i * NPAD + k0;
  float s = rhs[i];
#pragma unroll
  for (int p = 0; p < NB; ++p) s -= Mi[p] * y[p];
  rhs[i] = s;
}

__global__ void k_bwd_diag(const float* __restrict__ M, float* __restrict__ rhs, int k0) {
  __shared__ float y[NB];
  const int r = threadIdx.x;
  y[r] = rhs[k0 + r];
  __syncthreads();
  for (int p = NB - 1; p >= 0; --p) {
    if (r == p) y[r] = y[r] / M[(size_t)(k0 + p) * NPAD + k0 + p];
    __syncthreads();
    if (r < p) y[r] -= M[(size_t)(k0 + r) * NPAD + k0 + p] * y[p];
    __syncthreads();
  }
  rhs[k0 + r] = y[r];
}

__global__ void k_bwd_update(const float* __restrict__ M, float* __restrict__ rhs, int k0) {
  __shared__ float y[NB];
  const int t = threadIdx.x;
  if (t < NB) y[t] = rhs[k0 + t];
  __syncthreads();
  const int i = blockIdx.x * blockDim.x + t;
  if (i >= k0) return;
  const float* Mi = M + (size_t)i * NPAD + k0;
  float s = rhs[i];
#pragma unroll
  for (int p = 0; p < NB; ++p) s -= Mi[p] * y[p];
  rhs[i] = s;
}

__global__ void k_copy_out(const float* __restrict__ rhs, float* __restrict__ out) {
  const int i = blockIdx.x * blockDim.x + threadIdx.x;
  if (i < NV) out[i] = rhs[i];
}

// ---------------------------------------------------------------------------
// Host driver: deterministic sequence of launches on `stream`
// Workspace layout: M (NPAD*NPAD f32, ~146 MB, L2-resident) | rhs (NPAD f32)
// ---------------------------------------------------------------------------
extern "C" void kernel_launch(void* const* d_in, const int* in_sizes, int n_in,
                              void* d_out, int out_size, void* d_ws, size_t ws_size,
                              hipStream_t stream) {
  (void)in_sizes; (void)n_in; (void)out_size; (void)ws_size;
  const float* x_init = (const float*)d_in[0];
  const float* W      = (const float*)d_in[1];
  const float* x_nom  = (const float*)d_in[2];
  const float* A      = (const float*)d_in[3];
  const float* b0     = (const float*)d_in[4];
  // d_in[5] = G, d_in[6] = h : unused by the reference output

  float* M   = (float*)d_ws;
  float* rhs = M + (size_t)NPAD * NPAD;

  // 1) Zero matrix + rhs, build Q (WMMA), fill A/A^T blocks + padding diag, rhs
  const size_t ztot = (size_t)NPAD * NPAD + NPAD;
  k_zero<<<dim3((unsigned)((ztot + 255) / 256)), 256, 0, stream>>>(M, ztot);
  k_build_q<<<dim3((NV + 31) / 32, (NV + 31) / 32), 32, 0, stream>>>(W, M);
  const size_t nfill = (size_t)NEQ * NV + (size_t)(NPAD - NSYS);
  k_fill_kkt<<<dim3((unsigned)((nfill + 255) / 256)), 256, 0, stream>>>(A, M);
  k_rhs<<<dim3((NPAD + 255) / 256), 256, 0, stream>>>(M, x_nom, b0, x_init, rhs);

  // 2) Blocked LU (no pivoting): diag -> TRSMs -> WMMA trailing update
  for (int kb = 0; kb < NBLK; ++kb) {
    const int k0  = kb * NB;
    const int rem = NPAD - k0 - NB;
    k_lu_diag<<<1, NB, 0, stream>>>(M, k0);
    if (rem > 0) {
      k_trsm_left <<<dim3((rem + 255) / 256), 256, 0, stream>>>(M, k0);
      k_trsm_right<<<dim3((rem + 255) / 256), 256, 0, stream>>>(M, k0);
      k_lu_gemm   <<<dim3(rem / 32, rem / 32), 32, 0, stream>>>(M, k0);
    }
  }

  // 3) Forward substitution (L y = rhs)
  for (int kb = 0; kb < NBLK; ++kb) {
    const int k0  = kb * NB;
    const int rem = NPAD - k0 - NB;
    k_fwd_diag<<<1, NB, 0, stream>>>(M, rhs, k0);
    if (rem > 0) k_fwd_update<<<dim3((rem + 255) / 256), 256, 0, stream>>>(M, rhs, k0);
  }

  // 4) Backward substitution (U x = y)
  for (int kb = NBLK - 1; kb >= 0; --kb) {
    const int k0 = kb * NB;
    k_bwd_diag<<<1, NB, 0, stream>>>(M, rhs, k0);
    if (k0 > 0) k_bwd_update<<<dim3((k0 + 255) / 256), 256, 0, stream>>>(M, rhs, k0);
  }

  // 5) x_opt = sol[:NV]
  k_copy_out<<<dim3((NV + 255) / 256), 256, 0, stream>>>(rhs, (float*)d_out);
}